// QuantumStateProcessor_83863531421743
// MI455X (gfx1250) — compile-verified
//
#include <hip/hip_runtime.h>
#include <hip/hip_bf16.h>
#include <math.h>

#define N_QUBITS 14
#define N_STATE  (1 << N_QUBITS)          // 16384 complex amps (128 KB LDS)
#define N_PARAMS 140
#define BLOCK    512
#define NWAVES   (BLOCK / 32)             // 16 wave32 waves
#define NGROUPS3 (N_STATE / 8)            // 2048 groups of 8 amps
#define NCHUNKS3 (NGROUPS3 / 16)          // 128 chunks of 16 groups

typedef float v2f __attribute__((ext_vector_type(2)));
typedef float v8f __attribute__((ext_vector_type(8)));

struct cplx { float x, y; };

__device__ __forceinline__ cplx cmul(cplx a, cplx b) {
  return cplx{ a.x * b.x - a.y * b.y, a.x * b.y + a.y * b.x };
}

// Fused U = RZ(t3) * RY(t2) * RX(t1)  (state sees RX, RY, RZ in that order)
__device__ __forceinline__ void fused_u(float t1, float t2, float t3,
                                        cplx& u00, cplx& u01, cplx& u10, cplx& u11) {
  float s1, c1, s2, c2, s3, c3;
  __sincosf(0.5f * t1, &s1, &c1);
  __sincosf(0.5f * t2, &s2, &c2);
  __sincosf(0.5f * t3, &s3, &c3);
  cplx m00{  c2 * c1,  s2 * s1 };   // RY * RX
  cplx m01{ -s2 * c1, -c2 * s1 };
  cplx m10{  s2 * c1, -c2 * s1 };
  cplx m11{  c2 * c1, -s2 * s1 };
  cplx e0{ c3, -s3 }, e1{ c3, s3 }; // RZ = diag(e0, e1)
  u00 = cmul(e0, m00); u01 = cmul(e0, m01);
  u10 = cmul(e1, m10); u11 = cmul(e1, m11);
}

// Entry [r][c] of CRX embedded in the 8-dim (3-qubit-slot) space.
// ch = control slot, tg = target slot (slot s <-> amp-index bit s).
__device__ __forceinline__ cplx crx3(int r, int c, int ch, int tg, float ct, float st) {
  int diff = (r ^ c) & 7;
  int tb = 1 << tg;
  cplx out{ 0.f, 0.f };
  if ((diff & (7 ^ tb)) == 0) {         // only target bit may differ
    int ctrl = (r >> ch) & 1;
    if (!ctrl) { if (diff == 0) out.x = 1.f; }
    else       { if (diff == 0) out.x = ct; else out.y = -st; }
  }
  return out;
}

// Entry [r][c] of mat(second) * mat(first): row r of "second" has nonzeros
// only at columns r and r^(1<<tg2) (and is identity when its control bit is 0).
__device__ __forceinline__ cplx fused_crx(int r, int c,
                                          int ch1, int tg1, float ct1, float st1,
                                          int ch2, int tg2, float ct2, float st2) {
  cplx e = crx3(r, c, ch1, tg1, ct1, st1);
  int ctrl2 = (r >> ch2) & 1;
  if (!ctrl2) return e;
  cplx f = crx3(r ^ (1 << tg2), c, ch1, tg1, ct1, st1);
  // {ct2,0}*e + {0,-st2}*f
  return cplx{ ct2 * e.x + st2 * f.y, ct2 * e.y - st2 * f.x };
}

// Apply an 8x8 complex gate on bit positions p0 < p1 < p2 (slot s <-> bit ps).
// Real 16x16 form R3=[[X,-Y],[Y,X]], z=(re0..7, im0..7); split into four K=4
// slabs -> 4 chained V_WMMA_F32_16X16X4_F32, 16 groups (128 amps) per chain,
// 100% of rows useful. Lane needs the 4 complex entries
// u0=U3[rr][2hi], u1=U3[rr][2hi+1], u2=U3[rr][4+2hi], u3=U3[rr][5+2hi].
__device__ __forceinline__ void apply_triple(float2* st,
                                             cplx u0, cplx u1, cplx u2, cplx u3,
                                             int p0, int p1, int p2,
                                             int lane, int waveId) {
  const int  n    = lane & 15;
  const int  hi   = lane >> 4;
  const bool yrow = (n & 8) != 0;        // real rows 8..15 = [Y | X] half
  // A slabs (16x4): lane M=n, VGPR0 K=2*hi, VGPR1 K=2*hi+1
  const v2f A0 = { yrow ? u0.y : u0.x,  yrow ? u1.y : u1.x  };  // cols 0..3   [X;Y]
  const v2f A1 = { yrow ? u2.y : u2.x,  yrow ? u3.y : u3.x  };  // cols 4..7   [X;Y]
  const v2f A2 = { yrow ? u0.x : -u0.y, yrow ? u1.x : -u1.y };  // cols 8..11  [-Y;X]
  const v2f A3 = { yrow ? u2.x : -u2.y, yrow ? u3.x : -u3.y };  // cols 12..15 [-Y;X]
  const int b0 = 1 << p0, b1 = 1 << p1, b2 = 1 << p2;
  float* sf = reinterpret_cast<float*>(st);

  for (int chunk = waveId; chunk < NCHUNKS3; chunk += NWAVES) {
    int g = chunk * 16 + n;
    // insert zero bits at p0, then p1, then p2 (ascending)
    int l0 = g & (b0 - 1); int x    = ((g ^ l0) << 1) | l0;
    int l1 = x & (b1 - 1);     x    = ((x ^ l1) << 1) | l1;
    int l2 = x & (b2 - 1); int base = ((x ^ l2) << 1) | l2;
    // B slabs (4x16): lane N=n, VGPR0 K=2*hi, VGPR1 K=2*hi+1
    int    ih = base + (hi << p1);       // amp 2*hi
    float2 z0 = st[ih];                  // amp 2*hi
    float2 z1 = st[ih + b0];             // amp 2*hi+1
    float2 z2 = st[ih + b2];             // amp 4+2*hi
    float2 z3 = st[ih + b2 + b0];        // amp 5+2*hi
    v2f B0 = { z0.x, z1.x };             // re(amps 0..3)
    v2f B1 = { z2.x, z3.x };             // re(amps 4..7)
    v2f B2 = { z0.y, z1.y };             // im(amps 0..3)
    v2f B3 = { z2.y, z3.y };             // im(amps 4..7)
    v8f acc = { 0.f, 0.f, 0.f, 0.f, 0.f, 0.f, 0.f, 0.f };
    acc = __builtin_amdgcn_wmma_f32_16x16x4_f32(false, A0, false, B0, (short)0, acc, false, false);
    acc = __builtin_amdgcn_wmma_f32_16x16x4_f32(false, A1, false, B1, (short)0, acc, false, false);
    acc = __builtin_amdgcn_wmma_f32_16x16x4_f32(false, A2, false, B2, (short)0, acc, false, false);
    acc = __builtin_amdgcn_wmma_f32_16x16x4_f32(false, A3, false, B3, (short)0, acc, false, false);
    // D: lanes 0-15 rows 0..7 = re'(amps 0..7); lanes 16-31 rows 8..15 = im'.
    // Every lane stores 8 floats: float index 2*idx(amp) + hi. Full EXEC.
    sf[2 * base + hi]                     = acc[0];
    sf[2 * (base + b0) + hi]              = acc[1];
    sf[2 * (base + b1) + hi]              = acc[2];
    sf[2 * (base + b1 + b0) + hi]         = acc[3];
    sf[2 * (base + b2) + hi]              = acc[4];
    sf[2 * (base + b2 + b0) + hi]         = acc[5];
    sf[2 * (base + b2 + b1) + hi]         = acc[6];
    sf[2 * (base + b2 + b1 + b0) + hi]    = acc[7];
  }
}

// Fused pair of ring CRX gates: first CRX(qc1 -> qt1), then CRX(qt1 -> qt2).
__device__ __forceinline__ void crx_pair_apply(float2* st, int rr, int hi,
                                               int qc1, int qt1, int qt2,
                                               float th1, float th2,
                                               int lane, int waveId) {
  float s1, ct1, s2, ct2;
  __sincosf(0.5f * th1, &s1, &ct1);
  __sincosf(0.5f * th2, &s2, &ct2);
  int q0 = 13 - qc1, q1 = 13 - qt1, q2 = 13 - qt2;   // bit positions
  int pmax = q0 > q1 ? (q0 > q2 ? q0 : q2) : (q1 > q2 ? q1 : q2);
  int pmin = q0 < q1 ? (q0 < q2 ? q0 : q2) : (q1 < q2 ? q1 : q2);
  int pmid = q0 + q1 + q2 - pmax - pmin;
  int sc1 = (q0 == pmax) ? 2 : ((q0 == pmid) ? 1 : 0);  // slot of control 1
  int st1 = (q1 == pmax) ? 2 : ((q1 == pmid) ? 1 : 0);  // slot of target 1 == control 2
  int st2 = (q2 == pmax) ? 2 : ((q2 == pmid) ? 1 : 0);  // slot of target 2
  cplx u0 = fused_crx(rr, 2 * hi,     sc1, st1, ct1, s1, st1, st2, ct2, s2);
  cplx u1 = fused_crx(rr, 2 * hi + 1, sc1, st1, ct1, s1, st1, st2, ct2, s2);
  cplx u2 = fused_crx(rr, 4 + 2 * hi, sc1, st1, ct1, s1, st1, st2, ct2, s2);
  cplx u3 = fused_crx(rr, 5 + 2 * hi, sc1, st1, ct1, s1, st1, st2, ct2, s2);
  apply_triple(st, u0, u1, u2, u3, pmin, pmid, pmax, lane, waveId);
}

__global__ void __launch_bounds__(BLOCK)
qlcu_statevec_kernel(const float* __restrict__ params, float* __restrict__ out) {
  extern __shared__ float2 st[];           // 128 KB LDS-resident state
  const int b      = blockIdx.x;
  const int tid    = threadIdx.x;
  const int lane   = tid & 31;
  const int waveId = tid >> 5;
  const int n      = lane & 15;
  const int hi     = lane >> 4;
  const int rr     = n & 7;                // complex row of the 8x8 gate
  const float* __restrict__ p = params + (size_t)b * N_PARAMS;

  // |0...0>
  float4* st4 = reinterpret_cast<float4*>(st);
  for (int i = tid; i < N_STATE / 2; i += BLOCK)
    st4[i] = make_float4(0.f, 0.f, 0.f, 0.f);
  __syncthreads();
  if (tid == 0) st[0] = make_float2(1.f, 0.f);
  __syncthreads();

  int idx = 0;
  #pragma unroll 1
  for (int layer = 0; layer < 2; ++layer) {
    // ---- fused 1q gates: 4 triples (qubits 3t,3t+1,3t+2) + pair (12,13) ----
    #pragma unroll 1
    for (int t = 0; t < 5; ++t) {
      cplx a00, a01, a10, a11, b00, b01, b10, b11, c00, c01, c10, c11;
      int p0, p1, p2;
      if (t < 4) {
        fused_u(p[idx + 9*t + 0], p[idx + 9*t + 1], p[idx + 9*t + 2], a00, a01, a10, a11);
        fused_u(p[idx + 9*t + 3], p[idx + 9*t + 4], p[idx + 9*t + 5], b00, b01, b10, b11);
        fused_u(p[idx + 9*t + 6], p[idx + 9*t + 7], p[idx + 9*t + 8], c00, c01, c10, c11);
        p2 = 13 - 3*t; p1 = 12 - 3*t; p0 = 11 - 3*t;   // qubit 3t on high slot
      } else {
        // identity spectator on bit 2; qubit 12 -> bit 1, qubit 13 -> bit 0
        a00 = cplx{1.f, 0.f}; a01 = cplx{0.f, 0.f};
        a10 = cplx{0.f, 0.f}; a11 = cplx{1.f, 0.f};
        fused_u(p[idx + 36], p[idx + 37], p[idx + 38], b00, b01, b10, b11);
        fused_u(p[idx + 39], p[idx + 40], p[idx + 41], c00, c01, c10, c11);
        p2 = 2; p1 = 1; p0 = 0;
      }
      // U3[r][c] = Ua[r>>2][c>>2] * Ub[(r>>1)&1][(c>>1)&1] * Uc[r&1][c&1]
      cplx ua0 = (rr & 4) ? a10 : a00;                       // Ua[rr>>2][0]
      cplx ua1 = (rr & 4) ? a11 : a01;                       // Ua[rr>>2][1]
      cplx ubr = (rr & 2) ? (hi ? b11 : b10) : (hi ? b01 : b00); // Ub[(rr>>1)&1][hi]
      cplx uc0 = (rr & 1) ? c10 : c00;                       // Uc[rr&1][0]
      cplx uc1 = (rr & 1) ? c11 : c01;                       // Uc[rr&1][1]
      cplx w0 = cmul(ua0, ubr), w1 = cmul(ua1, ubr);
      cplx u0 = cmul(w0, uc0);   // col 2hi     (q=0,s=0)
      cplx u1 = cmul(w0, uc1);   // col 2hi+1
      cplx u2 = cmul(w1, uc0);   // col 4+2hi
      cplx u3 = cmul(w1, uc1);   // col 5+2hi
      apply_triple(st, u0, u1, u2, u3, p0, p1, p2, lane, waveId);
      __syncthreads();
    }
    idx += 42;

    // ---- forward CRX ring, fused in pairs: (G_2j, G_2j+1) on {2j,2j+1,2j+2} ----
    #pragma unroll 1
    for (int j = 0; j < 7; ++j) {
      int qc1 = 2*j, qt1 = 2*j + 1, qt2 = (2*j + 2) % N_QUBITS;
      crx_pair_apply(st, rr, hi, qc1, qt1, qt2,
                     p[idx + 2*j], p[idx + 2*j + 1], lane, waveId);
      __syncthreads();
    }
    idx += 14;

    // ---- backward CRX ring (i = 13..0), fused: (H_13-2k, H_12-2k) ----
    #pragma unroll 1
    for (int k = 0; k < 7; ++k) {
      int qc1 = 13 - 2*k, qt1 = 12 - 2*k, qt2 = (11 - 2*k + N_QUBITS) % N_QUBITS;
      crx_pair_apply(st, rr, hi, qc1, qt1, qt2,
                     p[idx + 2*k], p[idx + 2*k + 1], lane, waveId);
      __syncthreads();
    }
    idx += 14;
  }

  // complex64 out: interleaved (re, im) floats, coalesced float4 stores
  float4* o4 = reinterpret_cast<float4*>(out) + (size_t)b * (N_STATE / 2);
  for (int i = tid; i < N_STATE / 2; i += BLOCK)
    o4[i] = st4[i];
}

extern "C" void kernel_launch(void* const* d_in, const int* in_sizes, int n_in,
                              void* d_out, int out_size, void* d_ws, size_t ws_size,
                              hipStream_t stream) {
  (void)n_in; (void)out_size; (void)d_ws; (void)ws_size;
  const float* params = (const float*)d_in[0];
  float* out = (float*)d_out;
  int batch = in_sizes[0] / N_PARAMS;                    // 128
  size_t lds_bytes = (size_t)N_STATE * sizeof(float2);   // 128 KB of 320 KB/WGP
  hipFuncSetAttribute((const void*)qlcu_statevec_kernel,
                      hipFuncAttributeMaxDynamicSharedMemorySize, (int)lds_bytes);
  qlcu_statevec_kernel<<<batch, BLOCK, lds_bytes, stream>>>(params, out);
}